// AtomwiseReadout_56684978372798
// MI455X (gfx1250) — compile-verified
//
#include <hip/hip_runtime.h>
#include <hip/hip_bf16.h>

// CDNA5 / gfx1250. wave32. WMMA f32 16x16x4 used for the per-atom dot products
// (memory-bound kernel; matrix pipe is free).

typedef __attribute__((ext_vector_type(2))) float v2f;
typedef __attribute__((ext_vector_type(8))) float v8f;

#define FEAT 128
#define WAVES_PER_BLOCK 8

__global__ void AtomwiseReadout_zero_kernel(float* __restrict__ out, int n) {
    int i = blockIdx.x * blockDim.x + threadIdx.x;
    if (i < n) out[i] = 0.0f;
}

__global__ __launch_bounds__(256) void AtomwiseReadout_56684978372798_kernel(
    const float* __restrict__ f,       // [n_atoms, 128]
    const int*   __restrict__ seg,     // [n_atoms], sorted
    const float* __restrict__ w,       // [128]
    float*       __restrict__ out,     // [n_graphs], pre-zeroed
    int n_atoms, int n_tiles)
{
    const int lane = threadIdx.x & 31;
    const int wave = threadIdx.x >> 5;
    const int tile = blockIdx.x * WAVES_PER_BLOCK + wave;
    if (tile >= n_tiles) return;           // uniform per wave: EXEC stays all-1s below

    const int base = tile * 16;            // first atom of this wave's 16-row tile
    const int m    = lane & 15;            // row within tile
    const int half = lane >> 4;            // 0: K=k0..k0+1, 1: K=k0+2..k0+3 (A layout)

    int row = base + m;
    if (row >= n_atoms) row = n_atoms - 1; // clamp (dup read is safe; masked in epilogue)

    const float* fp = f + (size_t)row * FEAT + 2 * half;
    const float* wp = w + 2 * half;

    v8f c = {0.f, 0.f, 0.f, 0.f, 0.f, 0.f, 0.f, 0.f};

    // D = A(16x4) * B(4x16) + C over K=128 in 32 chained WMMAs.
    // B columns all equal w[k], so every column of D is the per-atom energy.
#pragma unroll
    for (int k0 = 0; k0 < FEAT; k0 += 4) {
        v2f a = *(const v2f*)(fp + k0);    // A: f[row][k0+2*half .. +1]
        v2f b = *(const v2f*)(wp + k0);    // B: w broadcast across N, same K split
        // (neg_a, A, neg_b, B, c_mod, C, reuse_a, reuse_b)
        c = __builtin_amdgcn_wmma_f32_16x16x4_f32(
                false, a, false, b, (short)0, c, false, false);
    }

    // C/D layout: VGPR r, lanes 0-15 -> M=r, lanes 16-31 -> M=r+8; all N identical.
    // Lane 0 owns atoms base..base+7, lane 16 owns base+8..base+15.
    if (m == 0) {
        const int a0 = base + 8 * half;
        float runsum = 0.0f;
        int   runseg = -1;
        for (int r = 0; r < 8; ++r) {
            const int atom = a0 + r;
            if (atom >= n_atoms) break;
            const int s = seg[atom];       // sorted -> long same-segment runs
            const float e = c[r];
            if (s == runseg) {
                runsum += e;
            } else {
                if (runseg >= 0) atomicAdd(&out[runseg], runsum);
                runseg = s;
                runsum = e;
            }
        }
        if (runseg >= 0) atomicAdd(&out[runseg], runsum);
    }
}

extern "C" void kernel_launch(void* const* d_in, const int* in_sizes, int n_in,
                              void* d_out, int out_size, void* d_ws, size_t ws_size,
                              hipStream_t stream) {
    // setup_inputs order: f [n_atoms*128] f32, segment_ids [n_atoms] i32,
    //                     n_graphs [1] i32 (device scalar; use out_size instead),
    //                     w_e [128] f32
    const float* f   = (const float*)d_in[0];
    const int*   seg = (const int*)d_in[1];
    const float* w   = (const float*)d_in[3];
    float*       out = (float*)d_out;

    const int n_atoms = in_sizes[0] / FEAT;
    const int n_tiles = (n_atoms + 15) / 16;

    // 1) zero the segment-sum output (harness poisons it)
    {
        const int threads = 256;
        const int blocks  = (out_size + threads - 1) / threads;
        AtomwiseReadout_zero_kernel<<<blocks, threads, 0, stream>>>(out, out_size);
    }

    // 2) fused dot + segment reduction: 8 waves/block, 16 atoms/wave
    {
        const int threads = 32 * WAVES_PER_BLOCK;
        const int blocks  = (n_tiles + WAVES_PER_BLOCK - 1) / WAVES_PER_BLOCK;
        AtomwiseReadout_56684978372798_kernel<<<blocks, threads, 0, stream>>>(
            f, seg, w, out, n_atoms, n_tiles);
    }
}